// Attention_65601330479503
// MI455X (gfx1250) — compile-verified
//
#include <hip/hip_runtime.h>
#include <math.h>

// ---------------- CDNA5 fp32 WMMA types ----------------
typedef __attribute__((ext_vector_type(2))) float v2f;   // A/B operand: 16x4 (or 4x16) f32 = 2 VGPRs
typedef __attribute__((ext_vector_type(8))) float v8f;   // C/D operand: 16x16 f32 = 8 VGPRs

#define DIMC   1024
#define HEADS  16
#define HD     64
#define NCLUST 8
#define BATCH  4
#define SEQ    1024
#define BHN    (BATCH*HEADS*SEQ)            // 65536

// d_out layout (floats): out | z_constraint | zero | zero | cp_bias
#define OUT_SZ (BATCH*SEQ*DIMC)             // 4194304
#define ZC_OFF (OUT_SZ)
#define CP_OFF (OUT_SZ + 3)

// workspace layout (floats)
#define WS_QKV   0                                   // [B*SEQ, 3*DIMC]
#define WS_ATTNO (WS_QKV + (size_t)BATCH*SEQ*3*DIMC) // [B*SEQ, DIMC]
#define WS_Z     (WS_ATTNO + (size_t)BATCH*SEQ*DIMC) // [B,H,SEQ,8]
#define WS_ZB    (WS_Z + (size_t)BHN*NCLUST)
#define WS_TH    (WS_ZB + (size_t)BHN*NCLUST)        // [B,H,SEQ]

static __device__ __forceinline__ v8f wmma4(v2f a, v2f b, v8f c) {
  // D = A(16x4,f32) x B(4x16,f32) + C(16x16,f32)
  return __builtin_amdgcn_wmma_f32_16x16x4_f32(false, a, false, b, (short)0, c, false, false);
}
static __device__ __forceinline__ v8f vzero8() {
  v8f v = {0.f,0.f,0.f,0.f,0.f,0.f,0.f,0.f};
  return v;
}
// reductions confined to 16-lane halves (wave32): masks 1,2,4,8 never cross half
static __device__ __forceinline__ float redmax16(float v) {
#pragma unroll
  for (int m = 1; m < 16; m <<= 1) v = fmaxf(v, __shfl_xor(v, m, 32));
  return v;
}
static __device__ __forceinline__ float redsum16(float v) {
#pragma unroll
  for (int m = 1; m < 16; m <<= 1) v += __shfl_xor(v, m, 32);
  return v;
}

// ---------------- generic fp32 WMMA GEMM: C[M,N] = A[M,K] @ B[K,N] (+bias) ----------------
// one wave (32 thr) computes a 64x64 tile of C (16 WMMA C-tiles); grid = (N/64, M/64)
// per K-chunk of 4: 4x A b64-loads + 8x B b32-loads feed 16 WMMAs.
// __launch_bounds__(32,1): single wave32 workgroup, min-occupancy register budget
// so the 128 accumulator VGPRs stay resident (no scratch spills).
__global__ void __launch_bounds__(32, 1)
gemm_f32_wmma(const float* __restrict__ A, const float* __restrict__ B,
              const float* __restrict__ bias, float* __restrict__ C,
              int K, int lda, int ldb, int ldc) {
  const int lane = threadIdx.x & 31;
  const int half = lane >> 4;
  const int lr   = lane & 15;
  const int n0 = blockIdx.x * 64;
  const int m0 = blockIdx.y * 64;

  v8f acc[4][4];
#pragma unroll
  for (int g = 0; g < 4; ++g)
#pragma unroll
    for (int t = 0; t < 4; ++t) acc[g][t] = vzero8();

  // A-layout: lane<16 holds (m, k=kc,kc+1); lane>=16 holds (m, k=kc+2,kc+3)
  const float* arow = A + (size_t)(m0 + lr) * lda + 2 * half;
  const size_t astride16 = (size_t)16 * lda;

  for (int kc = 0; kc < K; kc += 4) {
    v2f a[4];
#pragma unroll
    for (int g = 0; g < 4; ++g) a[g] = *(const v2f*)(arow + (size_t)g * astride16 + kc);
    // B-layout: lane<16 vgpr0/1 = B[kc+0/1, n]; lane>=16 = B[kc+2/3, n]
    const float* b0 = B + (size_t)(kc + 2 * half) * ldb + n0 + lr;
    const float* b1 = b0 + ldb;
#pragma unroll
    for (int t = 0; t < 4; ++t) {
      v2f bb; bb.x = b0[t * 16]; bb.y = b1[t * 16];
#pragma unroll
      for (int g = 0; g < 4; ++g) acc[g][t] = wmma4(a[g], bb, acc[g][t]);
    }
  }
  // C-layout store: vgpr r, lane -> (m = m0 + g*16 + r + 8*half, n = n0 + t*16 + lr)
#pragma unroll
  for (int t = 0; t < 4; ++t) {
    const int nc = n0 + t * 16 + lr;
    const float bv = bias ? bias[nc] : 0.0f;
#pragma unroll
    for (int g = 0; g < 4; ++g) {
#pragma unroll
      for (int r = 0; r < 8; ++r) {
        C[(size_t)(m0 + g * 16 + r + 8 * half) * ldc + nc] = acc[g][t][r] + bv;
      }
    }
  }
}

// ---------------- z / zB / theta / entropy ----------------
__global__ void zgroup_kernel(const float* __restrict__ qkv,
                              const float* __restrict__ Wgroup, const float* __restrict__ bgroup,
                              const float* __restrict__ Wtheta, const float* __restrict__ btheta,
                              const float* __restrict__ affB,
                              const int* __restrict__ flag, const int* __restrict__ cp_dcmm,
                              float* __restrict__ z, float* __restrict__ zB,
                              float* __restrict__ theta, float* __restrict__ zc_out) {
  if (!(flag[0] != 0 && cp_dcmm[0] != 0)) return;
  const int g = blockIdx.x * blockDim.x + threadIdx.x;      // [0, BHN)
  const int b = g >> 14;
  const int h = (g >> 10) & (HEADS - 1);
  const int n = g & (SEQ - 1);
  const float* q = qkv + (size_t)(b * SEQ + n) * (3 * DIMC) + h * HD;

  float logits[NCLUST];
#pragma unroll
  for (int l = 0; l < NCLUST; ++l) logits[l] = bgroup[l];
  float th = btheta[0];
  for (int d = 0; d < HD; ++d) {
    const float qd = q[d];
    th += qd * Wtheta[d];
#pragma unroll
    for (int l = 0; l < NCLUST; ++l) logits[l] += qd * Wgroup[d * NCLUST + l];
  }
  float mx = logits[0];
#pragma unroll
  for (int l = 1; l < NCLUST; ++l) mx = fmaxf(mx, logits[l]);
  float zv[NCLUST], se = 0.f;
#pragma unroll
  for (int l = 0; l < NCLUST; ++l) { zv[l] = __expf(logits[l] - mx); se += zv[l]; }
  const float inv = 1.0f / se;
  float ent = 0.f;
#pragma unroll
  for (int l = 0; l < NCLUST; ++l) { zv[l] *= inv; ent -= zv[l] * __logf(zv[l] + 1e-8f); }

  const size_t base = (size_t)g * NCLUST;
#pragma unroll
  for (int l2 = 0; l2 < NCLUST; ++l2) {
    float s = 0.f;
#pragma unroll
    for (int l = 0; l < NCLUST; ++l) {
      const float aff = affB[(h * NCLUST + l) * NCLUST + l2];
      s += zv[l] * (1.0f / (1.0f + __expf(-aff)));     // sigmoid
    }
    zB[base + l2] = s;
  }
#pragma unroll
  for (int l = 0; l < NCLUST; ++l) z[base + l] = zv[l];
  theta[g] = fmaxf(th, 0.0f);

  __shared__ float red[256];
  red[threadIdx.x] = ent * (1.0f / (float)BHN);
  __syncthreads();
  for (int s = 128; s > 0; s >>= 1) {
    if (threadIdx.x < s) red[threadIdx.x] += red[threadIdx.x + s];
    __syncthreads();
  }
  if (threadIdx.x == 0) atomicAdd(zc_out, red[0]);
}

// ---------------- fused flash attention + cluster-partition bias ----------------
// grid = (SEQ/32, HEADS, BATCH), 1 wave per block; each wave owns TWO stacked
// 16-row query tiles (32 rows) so every K-chunk / V-element load feeds 2 WMMAs.
// __launch_bounds__(32,1): lift the VGPR cap (Q tiles + accumulators ~ 280 regs).
__global__ void __launch_bounds__(32, 1)
flash_attn_cp_kernel(const float* __restrict__ qkv,
                     const float* __restrict__ z, const float* __restrict__ zB,
                     const float* __restrict__ theta,
                     const float* __restrict__ cp_scale_p,
                     const int* __restrict__ flag, const int* __restrict__ cp_dcmm,
                     float* __restrict__ attnout, float* __restrict__ cp_out) {
  const int lane = threadIdx.x & 31;
  const int half = lane >> 4;
  const int lr   = lane & 15;
  const int m0 = blockIdx.x * 32;
  const int h  = blockIdx.y;
  const int b  = blockIdx.z;
  const bool docp = (flag[0] != 0) && (cp_dcmm[0] != 0);
  const float cp_scale = cp_scale_p[0];
  const float sm_scale = 0.125f;                      // hd^-0.5 = 1/8
  const size_t rs = 3 * DIMC;                         // token row stride inside qkv
  const float* qbase = qkv + (size_t)b * SEQ * rs + h * HD;

  // Q tiles 2x(16x64) resident in registers, already in A-layout chunks
  v2f qa[2][16];
#pragma unroll
  for (int mt = 0; mt < 2; ++mt) {
    const float* qr = qbase + (size_t)(m0 + mt * 16 + lr) * rs + 2 * half;
#pragma unroll
    for (int c = 0; c < 16; ++c) qa[mt][c] = *(const v2f*)(qr + 4 * c);
  }

  const size_t bh = (size_t)(b * HEADS + h) * SEQ;
  v2f zba[2][2];
  float th_m[2][8];
#pragma unroll
  for (int mt = 0; mt < 2; ++mt) {
    zba[mt][0] = (v2f){0.f, 0.f};
    zba[mt][1] = (v2f){0.f, 0.f};
#pragma unroll
    for (int r = 0; r < 8; ++r) th_m[mt][r] = 0.f;
  }
  if (docp) {
#pragma unroll
    for (int mt = 0; mt < 2; ++mt) {
      const float* zr = zB + (bh + m0 + mt * 16 + lr) * NCLUST + 2 * half;
      zba[mt][0] = *(const v2f*)(zr);
      zba[mt][1] = *(const v2f*)(zr + 4);
#pragma unroll
      for (int r = 0; r < 8; ++r) th_m[mt][r] = theta[bh + m0 + mt * 16 + r + 8 * half];
    }
  }

  float mrun[2][8], lrun[2][8];
  v8f acc[2][4];
#pragma unroll
  for (int mt = 0; mt < 2; ++mt) {
#pragma unroll
    for (int r = 0; r < 8; ++r) { mrun[mt][r] = -1e30f; lrun[mt][r] = 0.f; }
#pragma unroll
    for (int t = 0; t < 4; ++t) acc[mt][t] = vzero8();
  }

  __shared__ float pl[2][16][18];                     // padded: even stride -> b64-aligned

  for (int n0 = 0; n0 < SEQ; n0 += 16) {
    // ---- S = Q @ K^T : each K chunk load feeds both m-tiles
    v8f s[2]; s[0] = vzero8(); s[1] = vzero8();
    const float* kr = qbase + DIMC + (size_t)(n0 + lr) * rs + 2 * half;
#pragma unroll
    for (int c = 0; c < 16; ++c) {
      v2f kb = *(const v2f*)(kr + 4 * c);
      s[0] = wmma4(qa[0][c], kb, s[0]);
      s[1] = wmma4(qa[1][c], kb, s[1]);
    }

    // ---- cp_bias tiles: tanh(zB_m . z_n) * theta_m * theta_n * scale
    v8f zz[2]; zz[0] = vzero8(); zz[1] = vzero8();
    float th_n = 0.f;
    if (docp) {
      const float* znr = z + (bh + n0 + lr) * NCLUST + 2 * half;
      v2f zb0 = *(const v2f*)(znr);
      v2f zb1 = *(const v2f*)(znr + 4);
#pragma unroll
      for (int mt = 0; mt < 2; ++mt) {
        zz[mt] = wmma4(zba[mt][0], zb0, zz[mt]);
        zz[mt] = wmma4(zba[mt][1], zb1, zz[mt]);
      }
      th_n = theta[bh + n0 + lr];
    }

    float pp[2][8];
#pragma unroll
    for (int mt = 0; mt < 2; ++mt) {
      float sa[8];
#pragma unroll
      for (int r = 0; r < 8; ++r) {
        const float cpb = docp ? (cp_scale * th_m[mt][r] * th_n * tanhf(zz[mt][r])) : 0.f;
        cp_out[(bh + m0 + mt * 16 + r + 8 * half) * SEQ + n0 + lr] = cpb;  // required output
        sa[r] = s[mt][r] * sm_scale + cpb;
      }
      // ---- online softmax (row m lives across one 16-lane half in vgpr r)
#pragma unroll
      for (int r = 0; r < 8; ++r) {
        const float tmax  = redmax16(sa[r]);
        const float nm    = fmaxf(mrun[mt][r], tmax);
        const float alpha = __expf(mrun[mt][r] - nm);
        const float p     = __expf(sa[r] - nm);
        lrun[mt][r] = lrun[mt][r] * alpha + redsum16(p);
        mrun[mt][r] = nm;
        pp[mt][r] = p;
#pragma unroll
        for (int t = 0; t < 4; ++t) acc[mt][t][r] *= alpha;
      }
    }

    // ---- transpose P tiles (C-layout -> A-layout) via LDS
    __syncthreads();
#pragma unroll
    for (int mt = 0; mt < 2; ++mt)
#pragma unroll
      for (int r = 0; r < 8; ++r) pl[mt][r + 8 * half][lr] = pp[mt][r];
    __syncthreads();
    v2f pa[2][4];
#pragma unroll
    for (int mt = 0; mt < 2; ++mt)
#pragma unroll
      for (int c = 0; c < 4; ++c) pa[mt][c] = *(const v2f*)(&pl[mt][lr][4 * c + 2 * half]);

    // ---- O += P @ V : every V element load feeds both m-tiles
    const float* vbase = qbase + 2 * DIMC + (size_t)n0 * rs;
#pragma unroll
    for (int c = 0; c < 4; ++c) {
      const float* v0 = vbase + (size_t)(4 * c + 2 * half) * rs + lr;
      const float* v1 = v0 + rs;
#pragma unroll
      for (int t = 0; t < 4; ++t) {
        v2f vb; vb.x = v0[t * 16]; vb.y = v1[t * 16];
        acc[0][t] = wmma4(pa[0][c], vb, acc[0][t]);
        acc[1][t] = wmma4(pa[1][c], vb, acc[1][t]);
      }
    }
  }

  // ---- normalize, store in [b, token, h*64+d] layout for the projection GEMM
#pragma unroll
  for (int mt = 0; mt < 2; ++mt) {
#pragma unroll
    for (int r = 0; r < 8; ++r) {
      const float inv = 1.0f / lrun[mt][r];
      const int m = m0 + mt * 16 + r + 8 * half;
#pragma unroll
      for (int t = 0; t < 4; ++t) {
        attnout[(size_t)(b * SEQ + m) * DIMC + h * HD + t * 16 + lr] = acc[mt][t][r] * inv;
      }
    }
  }
}

__global__ void init_scalars_kernel(float* __restrict__ p) {
  if (threadIdx.x < 3) p[threadIdx.x] = 0.0f;        // z_constraint + two zeros
}

// ---------------- launch ----------------
extern "C" void kernel_launch(void* const* d_in, const int* in_sizes, int n_in,
                              void* d_out, int out_size, void* d_ws, size_t ws_size,
                              hipStream_t stream) {
  const float* x       = (const float*)d_in[0];
  const float* Wqkv    = (const float*)d_in[1];
  const float* Wproj   = (const float*)d_in[2];
  const float* bproj   = (const float*)d_in[3];
  const float* Wgroup  = (const float*)d_in[4];
  const float* bgroup  = (const float*)d_in[5];
  const float* Wtheta  = (const float*)d_in[6];
  const float* btheta  = (const float*)d_in[7];
  const float* affB    = (const float*)d_in[8];
  const float* cpscale = (const float*)d_in[9];
  const int*   flag    = (const int*)d_in[10];
  const int*   cp_dcmm = (const int*)d_in[12];

  float* out = (float*)d_out;
  float* ws  = (float*)d_ws;
  float* qkv   = ws + WS_QKV;
  float* attno = ws + WS_ATTNO;
  float* zbuf  = ws + WS_Z;
  float* zBbuf = ws + WS_ZB;
  float* thbuf = ws + WS_TH;

  init_scalars_kernel<<<1, 32, 0, stream>>>(out + ZC_OFF);

  // qkv = x @ Wqkv : [4096 x 3072]
  gemm_f32_wmma<<<dim3(3 * DIMC / 64, BATCH * SEQ / 64), 32, 0, stream>>>(
      x, Wqkv, nullptr, qkv, DIMC, DIMC, 3 * DIMC, 3 * DIMC);

  // z / zB / theta / entropy
  zgroup_kernel<<<BHN / 256, 256, 0, stream>>>(
      qkv, Wgroup, bgroup, Wtheta, btheta, affB, flag, cp_dcmm,
      zbuf, zBbuf, thbuf, out + ZC_OFF);

  // fused attention + cp_bias (writes cp_bias output region directly)
  flash_attn_cp_kernel<<<dim3(SEQ / 32, HEADS, BATCH), 32, 0, stream>>>(
      qkv, zbuf, zBbuf, thbuf, cpscale, flag, cp_dcmm, attno, out + CP_OFF);

  // out = attno @ Wproj + bproj
  gemm_f32_wmma<<<dim3(DIMC / 64, BATCH * SEQ / 64), 32, 0, stream>>>(
      attno, Wproj, bproj, out, DIMC, DIMC, DIMC, DIMC);
}